// ScaledDotProductAttention2_64493228917085
// MI455X (gfx1250) — compile-verified
//
#include <hip/hip_runtime.h>
#include <math.h>

// MI455X / gfx1250, wave32. One workgroup = 8 waves = one batch group of 8 heads.
// All matmuls (QK^T, linear, head-mixing conv-as-GEMM, P@V) use
// V_WMMA_F32_16X16X4_F32: inputs are f32 and the kernel is HBM-bound
// (~550 MB -> ~24us at 23.3 TB/s), so full-precision f32 WMMA is free.

typedef float v2f __attribute__((ext_vector_type(2)));
typedef float v8f __attribute__((ext_vector_type(8)));

#define L   27
#define LP  28      // padded LDS row stride
#define D   64
#define NH  8
#define BHN 16384
#define M2  216     // NH*L: conv-as-GEMM dimension

// Constant conv GEMM matrix, identical for every block: 186 KB, L2-resident.
// g_W2[o*27+h][q'*8+i] = (q' <= h) ? conv_w[o][i][q'-h+26] : 0
__device__ float g_W2[M2 * M2];

__global__ __launch_bounds__(256)
void build_w2(const float* __restrict__ cw) {
  int idx = blockIdx.x * 256 + threadIdx.x;
  if (idx >= M2 * M2) return;
  int row = idx / M2;            // o*27 + h
  int kk  = idx - row * M2;      // q'*8 + i
  int o  = row / L, h = row - o * L;
  int qp = kk >> 3, i = kk & 7;
  float val = 0.f;
  if (qp <= h) val = cw[(o * NH + i) * L + (qp - h + 26)];
  g_W2[idx] = val;
}

__device__ __forceinline__ v8f wmma4(v2f a, v2f b, v8f c) {
  // D = A(16x4 f32) * B(4x16 f32) + C(16x16 f32)
  return __builtin_amdgcn_wmma_f32_16x16x4_f32(false, a, false, b, (short)0, c,
                                               false, false);
}

__global__ __launch_bounds__(256)
void attn_fused(const float* __restrict__ q,
                const float* __restrict__ k,
                const float* __restrict__ v,
                const float* __restrict__ ap,
                const float* __restrict__ lw,
                const float* __restrict__ lb,
                float* __restrict__ outp,
                float* __restrict__ attn_out)
{
  const int tid  = threadIdx.x;
  const int lane = tid & 31;
  const int wid  = tid >> 5;          // head 0..7, one wave per head
  const int l16  = lane & 15;
  const int lh   = lane >> 4;         // half-wave select
  const int bh   = blockIdx.x * NH + wid;

  __shared__ float sS[NH * LP * LP];  // blended scores -> probs
  __shared__ float sC[NH * LP * LP];  // linear output = conv-GEMM input X[(q',i)][w]

  const float* qh  = q  + (size_t)bh * (L * D);
  const float* kh  = k  + (size_t)bh * (L * D);
  const float* vh  = v  + (size_t)bh * (L * D);
  const float* aph = ap + (size_t)bh * (L * L);
  float* sSh = sS + wid * LP * LP;
  float* sCh = sC + wid * LP * LP;

  const v8f vz = {0.f,0.f,0.f,0.f,0.f,0.f,0.f,0.f};

  // ---------- Phase 1: S = (q @ k^T)/8, blended 0.5*pre + 0.5*S ----------
  // Tile (m<16, n>=16) is fully masked downstream -> skipped.
  // Row/col clamp to 26: clamped lanes only feed discarded C rows/cols.
  const int r0 = l16;
  const int r1 = (16 + l16 > 26) ? 26 : (16 + l16);
  v8f s00 = vz, s10 = vz, s11 = vz;
  #pragma unroll
  for (int kb = 0; kb < D; kb += 4) {
    const int kc = kb + 2 * lh;                 // A: K=kc,kc+1 per half-wave
    v2f a0 = *(const v2f*)(qh + r0 * D + kc);
    v2f a1 = *(const v2f*)(qh + r1 * D + kc);
    v2f b0 = *(const v2f*)(kh + r0 * D + kc);   // B = k^T: B[kc][n] = k[n][kc]
    v2f b1 = *(const v2f*)(kh + r1 * D + kc);
    s00 = wmma4(a0, b0, s00);
    s10 = wmma4(a1, b0, s10);
    s11 = wmma4(a1, b1, s11);
  }
  #pragma unroll
  for (int r = 0; r < 8; ++r) {
    { int m = r + 8 * lh, n = l16;              // tile (0,0)
      if (m < 16) sSh[m * LP + n] = 0.5f * aph[m * L + n] + 0.0625f * s00[r]; }
    { int m = 16 + r + 8 * lh, n = l16;         // tile (1,0)
      if (m < L) sSh[m * LP + n] = 0.5f * aph[m * L + n] + 0.0625f * s10[r]; }
    { int m = 16 + r + 8 * lh, n = 16 + l16;    // tile (1,1)
      if (m < L && n < L)
        sSh[m * LP + n] = 0.5f * aph[m * L + n] + 0.0625f * s11[r]; }
  }

  // ---------- Phase 2: attn_cnn = masked(S) @ lin_w^T + b ----------
  // masked(S) rows m<=15 are zero for k>=16 -> mt0 only runs kb<16.
  v8f accL[2][2] = {{vz, vz}, {vz, vz}};
  #pragma unroll
  for (int kb = 0; kb < LP; kb += 4) {          // K padded 27 -> 28
    const int kc = kb + 2 * lh;
    v2f bf[2];
    #pragma unroll
    for (int nt = 0; nt < 2; ++nt) {
      int j = nt * 16 + l16;                    // B[kc][j] = lin_w[j][kc]
      bool jv = j < L;
      bf[nt].x = (jv && kc     < L) ? lw[j * L + kc]     : 0.f;
      bf[nt].y = (jv && kc + 1 < L) ? lw[j * L + kc + 1] : 0.f;
    }
    {                                            // mt = 1
      int m = 16 + l16, mc = m > 26 ? 26 : m;
      float ax = sSh[mc * LP + kc], ay = sSh[mc * LP + kc + 1];
      v2f a;
      a.x = (kc     <= m && kc     < L) ? ax : 0.f;   // causal mask on the fly
      a.y = (kc + 1 <= m && kc + 1 < L) ? ay : 0.f;
      accL[1][0] = wmma4(a, bf[0], accL[1][0]);
      accL[1][1] = wmma4(a, bf[1], accL[1][1]);
    }
    if (kb < 16) {                               // mt = 0
      int m = l16;
      float ax = sSh[m * LP + kc], ay = sSh[m * LP + kc + 1];
      v2f a;
      a.x = (kc     <= m) ? ax : 0.f;
      a.y = (kc + 1 <= m) ? ay : 0.f;
      accL[0][0] = wmma4(a, bf[0], accL[0][0]);
      accL[0][1] = wmma4(a, bf[1], accL[0][1]);
    }
  }
  #pragma unroll
  for (int mt = 0; mt < 2; ++mt)
    #pragma unroll
    for (int nt = 0; nt < 2; ++nt)
      #pragma unroll
      for (int r = 0; r < 8; ++r) {
        int m = mt * 16 + r + 8 * lh;
        int n = nt * 16 + l16;
        if (m < L && n < L)
          sCh[m * LP + n] = accL[mt][nt][r] + lb[n];
      }

  __syncthreads();   // conv-GEMM mixes all 8 heads' sC

  // ---------- Phase 3: causal head-mixing conv as GEMM vs constant W2 ----
  // Y[h][w] (head o) = sum_k W2[o*27+h][k] * X[k][w], X[(q'*8+i)][w]=sC[i][q'][w]
  // Rows h<=15 need q'<=15 only -> mt0 stops at k=128.
  {
    const int o = wid;
    v8f accY[2][2] = {{vz, vz}, {vz, vz}};
    const float* w2r0 = g_W2 + (o * L + l16) * M2 + 2 * lh;
    int row1 = o * L + 16 + l16; if (row1 > M2 - 1) row1 = M2 - 1;
    const float* w2r1 = g_W2 + row1 * M2 + 2 * lh;
    #pragma unroll
    for (int kb = 0; kb < M2; kb += 4) {
      const int qp = kb >> 3;                // same q' for both half-waves
      const int ib = kb & 7;                 // i = ib + 2*lh, pair uses i, i+1
      v2f bf[2];
      #pragma unroll
      for (int nt = 0; nt < 2; ++nt) {
        const float* xb = sC + (ib + 2 * lh) * (LP * LP) + qp * LP + nt * 16 + l16;
        bf[nt].x = xb[0];
        bf[nt].y = xb[LP * LP];
      }
      v2f a1 = *(const v2f*)(w2r1 + kb);
      accY[1][0] = wmma4(a1, bf[0], accY[1][0]);
      accY[1][1] = wmma4(a1, bf[1], accY[1][1]);
      if (kb < 128) {
        v2f a0 = *(const v2f*)(w2r0 + kb);
        accY[0][0] = wmma4(a0, bf[0], accY[0][0]);
        accY[0][1] = wmma4(a0, bf[1], accY[0][1]);
      }
    }
    // relu + blend: attn = 0.1*relu(y) + 0.9*attn
    #pragma unroll
    for (int mt = 0; mt < 2; ++mt)
      #pragma unroll
      for (int nt = 0; nt < 2; ++nt)
        #pragma unroll
        for (int r = 0; r < 8; ++r) {
          int h = mt * 16 + r + 8 * lh;
          int w = nt * 16 + l16;
          if (h < L && w < L) {
            float yv = accY[mt][nt][r];
            yv = yv > 0.f ? yv : 0.f;
            sSh[h * LP + w] = 0.1f * yv + 0.9f * sSh[h * LP + w];
          }
        }
  }

  // ---------- Phase 4: causal softmax over keys (row per lane) ----------
  if (lane < L) {
    const int m = lane;
    float* row = sSh + m * LP;
    float mx = -3.402823466e38f;
    for (int n = 0; n <= m; ++n) mx = fmaxf(mx, row[n]);
    float sum = 0.f;
    for (int n = 0; n <= m; ++n) { float p = __expf(row[n] - mx); sum += p; row[n] = p; }
    float rs = 1.0f / sum;
    for (int n = 0; n <= m; ++n) row[n] *= rs;
    for (int n = m + 1; n < LP; ++n) row[n] = 0.f;   // mask + K-pad col 27 = 0
  }

  // Coalesced attn writeback
  {
    float* aout = attn_out + (size_t)bh * (L * L);
    if (lane < L)
      for (int m = 0; m < L; ++m)
        aout[m * L + lane] = sSh[m * LP + lane];
  }

  // ---------- Phase 5: O = P @ V ----------
  // Prob rows m<=15 are zero for k>=16 -> mt0 only runs kb<16.
  v8f accO[2][4] = {{vz, vz, vz, vz}, {vz, vz, vz, vz}};
  #pragma unroll
  for (int kb = 0; kb < LP; kb += 4) {
    const int kc = kb + 2 * lh;
    const int k0 = kc     > 26 ? 26 : kc;    // A col 27 is 0 -> B row value moot
    const int k1 = kc + 1 > 26 ? 26 : kc + 1;
    v2f a1;
    { int m = 16 + l16, mc = m > 26 ? 26 : m;
      a1.x = sSh[mc * LP + kc]; a1.y = sSh[mc * LP + kc + 1]; }
    v2f a0m = {0.f, 0.f};
    if (kb < 16) { int m = l16;
      a0m.x = sSh[m * LP + kc]; a0m.y = sSh[m * LP + kc + 1]; }
    #pragma unroll
    for (int nt = 0; nt < 4; ++nt) {
      int n = nt * 16 + l16;
      v2f bfv;
      bfv.x = vh[k0 * D + n];
      bfv.y = vh[k1 * D + n];
      accO[1][nt] = wmma4(a1, bfv, accO[1][nt]);
      if (kb < 16) accO[0][nt] = wmma4(a0m, bfv, accO[0][nt]);
    }
  }
  {
    float* oh = outp + (size_t)bh * (L * D);
    #pragma unroll
    for (int mt = 0; mt < 2; ++mt)
      #pragma unroll
      for (int nt = 0; nt < 4; ++nt)
        #pragma unroll
        for (int r = 0; r < 8; ++r) {
          int m = mt * 16 + r + 8 * lh;
          int n = nt * 16 + l16;
          if (m < L)
            oh[m * D + n] = accO[mt][nt][r];
        }
  }
}

extern "C" void kernel_launch(void* const* d_in, const int* in_sizes, int n_in,
                              void* d_out, int out_size, void* d_ws, size_t ws_size,
                              hipStream_t stream) {
  const float* q  = (const float*)d_in[0];
  const float* k  = (const float*)d_in[1];
  const float* v  = (const float*)d_in[2];
  const float* ap = (const float*)d_in[3];
  // d_in[4] is the bool causal mask: a fixed triu(k=1) pattern, recomputed
  // arithmetically in-kernel (k > q) to avoid bool-ABI ambiguity.
  const float* lw = (const float*)d_in[5];
  const float* lb = (const float*)d_in[6];
  const float* cw = (const float*)d_in[7];

  float* outp = (float*)d_out;                       // (BH, L, D) f32
  float* attn = outp + (size_t)BHN * L * D;          // (BH, L, L) f32

  // Rebuild the constant conv-GEMM matrix every launch (deterministic).
  build_w2<<<(M2 * M2 + 255) / 256, 256, 0, stream>>>(cw);
  attn_fused<<<BHN / NH, 256, 0, stream>>>(q, k, v, ap, lw, lb, outp, attn);
}